// ConstraintModule_77575699301004
// MI455X (gfx1250) — compile-verified
//
#include <hip/hip_runtime.h>
#include <math.h>

// Problem constants (fixed by the reference harness)
#define B_TOT   32768
#define IN_DIM  256
#define NDIM    128
#define KDIM    128
#define MLIN    1024
#define QC      16

// Tiling: 64 batch rows per workgroup, 512 threads = 16 wave32.
#define BT      64
#define THREADS 512
#define NWAVES  16
#define VLD     (NDIM + 4)   // 132: stride mod 64 banks = 4 -> conflict-free frags
#define XLD     (IN_DIM + 4) // 260: same property for the x tile

typedef __attribute__((ext_vector_type(2))) float v2f;
typedef __attribute__((ext_vector_type(8))) float v8f;

// A fragment (16x4 f32, MxK): lanes 0-15 rows M, VGPR0/1 = K0/K1 (lanes<16) or K2/K3.
static __device__ __forceinline__ v2f frag_a(const float* p, int ld, int lane) {
  const int row = lane & 15;
  const int kb  = (lane >> 4) << 1;
  v2f a;
  a.x = p[row * ld + kb];
  a.y = p[row * ld + kb + 1];
  return a;
}

// B fragment (4x16 f32, KxN) from a row-major [ncols x ld] matrix: B[k][n] = p[n*ld+k].
static __device__ __forceinline__ v2f frag_bT(const float* p, int ld, int lane) {
  const int col = lane & 15;
  const int kr  = (lane >> 4) << 1;
  v2f b;
  b.x = p[col * ld + kr];
  b.y = p[col * ld + kr + 1];
  return b;
}

static __device__ __forceinline__ v8f wmma4(v2f a, v2f b, v8f c) {
  return __builtin_amdgcn_wmma_f32_16x16x4_f32(false, a, false, b, (short)0, c,
                                               false, false);
}

// One-shot setup: constk = NA_E @ z0 + yp  (block-invariant, hoisted out)
__global__ void constk_init(const float* __restrict__ NA_E,
                            const float* __restrict__ yp,
                            const float* __restrict__ z0,
                            float* __restrict__ constk) {
  const int k = threadIdx.x;           // 128 threads
  float s = yp[k];
  for (int n = 0; n < NDIM; ++n) s += NA_E[k * NDIM + n] * z0[n];
  constk[k] = s;
}

// LDS plan (floats):
//   vbuf [BT][VLD]                          8448
//   rbuf [BT][VLD]                          8448   (xtile[BT][XLD]=16640 aliases vbuf+rbuf)
//   partQ[16][BT]                           1024
//   partL[16][BT]                           1024
//   norms[BT] klin[BT] alphas[BT]            192
//   constk[KDIM]                             128
#define SMEM_FLOATS (2 * BT * VLD + 2 * NWAVES * BT + 3 * BT + KDIM)

__global__ __launch_bounds__(THREADS)
void rayen_fused(const float* __restrict__ x,     // [B, 256]
                 const float* __restrict__ W,     // [128, 256]
                 const float* __restrict__ bias,  // [128]
                 const float* __restrict__ D,     // [1024, 128]
                 const float* __restrict__ NA_E,  // [128, 128]
                 const float* __restrict__ yp,    // [128]
                 const float* __restrict__ z0,    // [128]
                 const float* __restrict__ phi,   // [16, 128]
                 const float* __restrict__ delta, // [16, 128, 128]
                 const float* __restrict__ constk_g, // [128] or null
                 float* __restrict__ y)           // [B, 128]
{
  __shared__ __align__(16) float smem[SMEM_FLOATS];
  float* vbuf   = smem;                     // [BT][VLD]
  float* rbuf   = vbuf + BT * VLD;          // [BT][VLD]
  float* xtile  = vbuf;                     // [BT][XLD], aliases vbuf+rbuf
  float* partQ  = rbuf + BT * VLD;          // [NWAVES][BT]
  float* partL  = partQ + NWAVES * BT;      // [NWAVES][BT]
  float* norms  = partL + NWAVES * BT;      // [BT]
  float* klin   = norms + BT;               // [BT]
  float* alphas = klin + BT;                // [BT]
  float* constk = alphas + BT;              // [KDIM]

  const int tid  = threadIdx.x;
  const int wave = tid >> 5;
  const int lane = tid & 31;
  const int half = lane >> 4;
  const int col  = lane & 15;
  const int cw   = wave & 7;                // stages 1/4/5: column-tile owner
  const int rg   = wave >> 3;               // stages 1/4: row pair; stage 5: K half
  const int bbase = blockIdx.x * BT;

  if (tid < KDIM) {
    if (constk_g) {
      constk[tid] = constk_g[tid];
    } else {
      float s = yp[tid];
      for (int n = 0; n < NDIM; ++n) s += NA_E[tid * NDIM + n] * z0[n];
      constk[tid] = s;
    }
  }

  // ---- Stage 0: coalesced float4 load of x tile [64,256] into LDS ----------
  #pragma unroll
  for (int it = 0; it < (BT * IN_DIM / 4) / THREADS; ++it) {   // 8 iters
    const int g   = it * THREADS + tid;       // float4 index
    const int row = g >> 6;                   // 64 float4 per row
    const int c4  = g & 63;
    const float4 v4 = *(const float4*)(x + (size_t)(bbase + row) * IN_DIM + c4 * 4);
    *(float4*)(xtile + row * XLD + c4 * 4) = v4;
  }
  __syncthreads();

  // ---- Stage 1: v = x @ W^T + b  ([BT,256] x [256,128]) --------------------
  {
    v8f acc0 = {}, acc1 = {};
    const float* bp = W + (size_t)(cw * 16) * IN_DIM;
    #pragma unroll 4
    for (int kk = 0; kk < IN_DIM; kk += 4) {
      const v2f bf = frag_bT(bp + kk, IN_DIM, lane);            // shared by both rt
      acc0 = wmma4(frag_a(xtile + ((rg * 2 + 0) * 16) * XLD + kk, XLD, lane), bf, acc0);
      acc1 = wmma4(frag_a(xtile + ((rg * 2 + 1) * 16) * XLD + kk, XLD, lane), bf, acc1);
    }
    __syncthreads();                       // xtile reads done before vbuf overwrite
    const int cc = cw * 16 + col;
    const float bv = bias[cc];
    const int r0 = (rg * 2 + 0) * 16 + half * 8;
    const int r1 = (rg * 2 + 1) * 16 + half * 8;
    #pragma unroll
    for (int r = 0; r < 8; ++r) {
      vbuf[(r0 + r) * VLD + cc] = acc0[r] + bv;
      vbuf[(r1 + r) * VLD + cc] = acc1[r] + bv;
    }
  }
  __syncthreads();

  // ---- Stage 2: row norms + normalize --------------------------------------
  for (int rr = 0; rr < BT / NWAVES; ++rr) {
    const int row = wave * (BT / NWAVES) + rr;
    float s = 0.f;
    for (int c = lane; c < NDIM; c += 32) { const float t = vbuf[row * VLD + c]; s += t * t; }
    #pragma unroll
    for (int mk = 16; mk >= 1; mk >>= 1) s += __shfl_xor(s, mk, 32);
    if (lane == 0) norms[row] = sqrtf(s);
  }
  __syncthreads();
  {
    const int row = tid >> 3;
    const int c0  = (tid & 7) * 16;
    const float inv = 1.0f / fmaxf(norms[row], 1e-12f);
    #pragma unroll
    for (int c = 0; c < 16; ++c) vbuf[row * VLD + c0 + c] *= inv;
  }
  __syncthreads();

  // ---- Stage 3: kappa_lin = relu(max_m (D v_bar)) --------------------------
  // Each wave owns 4 UNIQUE column tiles (ct = wave + 16*ci) and computes all
  // 4 row tiles per B fragment -> every D element read exactly once per block.
  {
    float rmax[4][8];
    #pragma unroll
    for (int t = 0; t < 4; ++t)
      #pragma unroll
      for (int r = 0; r < 8; ++r) rmax[t][r] = -3.402823466e38f;

    for (int ci = 0; ci < MLIN / 16 / NWAVES; ++ci) {   // 4 iterations
      const int ct = wave + ci * NWAVES;
      const float* bp = D + (size_t)(ct * 16) * NDIM;
      v8f acc[4] = {{}, {}, {}, {}};
      #pragma unroll 2
      for (int kk = 0; kk < NDIM; kk += 4) {
        const v2f bf = frag_bT(bp + kk, NDIM, lane);    // shared by 4 row tiles
        #pragma unroll
        for (int t = 0; t < 4; ++t)
          acc[t] = wmma4(frag_a(vbuf + (t * 16) * VLD + kk, VLD, lane), bf, acc[t]);
      }
      #pragma unroll
      for (int t = 0; t < 4; ++t)
        #pragma unroll
        for (int r = 0; r < 8; ++r) rmax[t][r] = fmaxf(rmax[t][r], acc[t][r]);
    }
    #pragma unroll
    for (int t = 0; t < 4; ++t)
      #pragma unroll
      for (int r = 0; r < 8; ++r) {
        float m = rmax[t][r];
        #pragma unroll
        for (int mk = 1; mk < 16; mk <<= 1) m = fmaxf(m, __shfl_xor(m, mk, 32));
        if (col == 0) partQ[wave * BT + t * 16 + half * 8 + r] = m;
      }
  }
  __syncthreads();
  if (tid < BT) {
    float m = 0.f;                            // relu folded into init
    for (int w = 0; w < NWAVES; ++w) m = fmaxf(m, partQ[w * BT + tid]);
    klin[tid] = m;
  }
  __syncthreads();

  // ---- Stage 4: rho = NA_E @ v_bar -----------------------------------------
  {
    v8f acc0 = {}, acc1 = {};
    const float* bp = NA_E + (size_t)(cw * 16) * NDIM;
    #pragma unroll 4
    for (int kk = 0; kk < NDIM; kk += 4) {
      const v2f bf = frag_bT(bp + kk, NDIM, lane);
      acc0 = wmma4(frag_a(vbuf + ((rg * 2 + 0) * 16) * VLD + kk, VLD, lane), bf, acc0);
      acc1 = wmma4(frag_a(vbuf + ((rg * 2 + 1) * 16) * VLD + kk, VLD, lane), bf, acc1);
    }
    const int cc = cw * 16 + col;
    const int r0 = (rg * 2 + 0) * 16 + half * 8;
    const int r1 = (rg * 2 + 1) * 16 + half * 8;
    #pragma unroll
    for (int r = 0; r < 8; ++r) {
      rbuf[(r0 + r) * VLD + cc] = acc0[r];
      rbuf[(r1 + r) * VLD + cc] = acc1[r];
    }
  }
  __syncthreads();

  // ---- Stage 5: kappa_nl = max_q (phi_q.rho + sqrt(rho' delta_q rho)) ------
  // K-split: wave (cw, rg) handles ct=cw over k in [rg*64, rg*64+64), all 4
  // row tiles -> every delta element read exactly once per block. Partial
  // column-sums combine in the cross-wave LDS reduction.
  float knl = -3.402823466e38f;               // meaningful for tid < BT
  for (int q = 0; q < QC; ++q) {
    const float* bp  = delta + (size_t)q * KDIM * KDIM + (size_t)(cw * 16) * KDIM;
    const float  phc = phi[q * KDIM + cw * 16 + col];
    v8f acc[4] = {{}, {}, {}, {}};
    const int k0 = rg * (KDIM / 2);
    #pragma unroll 2
    for (int kk = k0; kk < k0 + KDIM / 2; kk += 4) {
      const v2f bf = frag_bT(bp + kk, KDIM, lane);      // shared by 4 row tiles
      #pragma unroll
      for (int t = 0; t < 4; ++t)
        acc[t] = wmma4(frag_a(rbuf + (t * 16) * VLD + kk, VLD, lane), bf, acc[t]);
    }
    #pragma unroll
    for (int t = 0; t < 4; ++t)
      #pragma unroll
      for (int r = 0; r < 8; ++r) {
        const int row = t * 16 + half * 8 + r;
        const float rv = rbuf[row * VLD + cw * 16 + col];
        float qs = acc[t][r] * rv;            // partial of rho' delta rho
        float ls = (rg == 0) ? phc * rv : 0.f;// phi.rho counted once per column
        #pragma unroll
        for (int mk = 1; mk < 16; mk <<= 1) {
          qs += __shfl_xor(qs, mk, 32);
          ls += __shfl_xor(ls, mk, 32);
        }
        if (col == 0) { partQ[wave * BT + row] = qs; partL[wave * BT + row] = ls; }
      }
    __syncthreads();
    if (tid < BT) {
      float sq = 0.f, sl = 0.f;
      for (int w = 0; w < NWAVES; ++w) { sq += partQ[w * BT + tid]; sl += partL[w * BT + tid]; }
      knl = fmaxf(knl, sl + sqrtf(fmaxf(sq, 0.f)));
    }
    __syncthreads();
  }

  // ---- Stage 6: alpha, y ----------------------------------------------------
  if (tid < BT) {
    const float kap = fmaxf(klin[tid], knl);       // klin >= 0 (relu)
    alphas[tid] = fminf(1.0f / kap, norms[tid]);   // kap==0 -> inf -> ||v||
  }
  __syncthreads();
  {
    const int row = tid >> 3;
    const int c0  = (tid & 7) * 16;
    const float a = alphas[row];
    float* yo = y + (size_t)(bbase + row) * KDIM;
    #pragma unroll
    for (int c = 0; c < 16; ++c)
      yo[c0 + c] = a * rbuf[row * VLD + c0 + c] + constk[c0 + c];
  }
}

extern "C" void kernel_launch(void* const* d_in, const int* in_sizes, int n_in,
                              void* d_out, int out_size, void* d_ws, size_t ws_size,
                              hipStream_t stream) {
  (void)in_sizes; (void)n_in; (void)out_size;
  const float* x     = (const float*)d_in[0];
  const float* W     = (const float*)d_in[1];
  const float* bias  = (const float*)d_in[2];
  const float* D     = (const float*)d_in[3];
  const float* NA_E  = (const float*)d_in[4];
  const float* yp    = (const float*)d_in[5];
  const float* z0    = (const float*)d_in[6];
  const float* phi   = (const float*)d_in[7];
  const float* delta = (const float*)d_in[8];
  float* y = (float*)d_out;

  const float* constk_g = nullptr;
  if (ws_size >= KDIM * sizeof(float)) {
    float* ck = (float*)d_ws;
    hipLaunchKernelGGL(constk_init, dim3(1), dim3(KDIM), 0, stream, NA_E, yp, z0, ck);
    constk_g = ck;
  }

  dim3 grid(B_TOT / BT), block(THREADS);
  hipLaunchKernelGGL(rayen_fused, grid, block, 0, stream,
                     x, W, bias, D, NA_E, yp, z0, phi, delta, constk_g, y);
}